// OuterProductMean_16655883174277
// MI455X (gfx1250) — compile-verified
//
#include <hip/hip_runtime.h>

// OuterProductMean, MI455X (gfx1250, wave32).
// Problem: msa(256,1024,256) f32 -> pair(1024,1024,128) f32.
// HBM-bound: 256MB in + 512MB out => ~33us floor @23.3TB/s; compute ~5 GFLOP.
// fp32 precision kept end-to-end via V_WMMA_F32_16X16X4_F32.

#define NSEQ 256
#define NRES 1024
#define CM   256
#define CC   32
#define CZ   128

#define XPAD 260   // 256+4: column reads across rows hit distinct banks (260%64==4)
#define APAD 44    // 8*44%64==32 (D-write halves disjoint), 2*44%64==24 (K/K+2 reads disjoint)

typedef __attribute__((ext_vector_type(2))) float v2f;
typedef __attribute__((ext_vector_type(4))) float v4f;
typedef __attribute__((ext_vector_type(8))) float v8f;

__global__ __launch_bounds__(128)
void opm_fused_kernel(const float* __restrict__ msa,
                      const float* __restrict__ norm_w,
                      const float* __restrict__ norm_b,
                      const float* __restrict__ w_ab,
                      const float* __restrict__ b_ab,
                      const float* __restrict__ w_out,
                      const float* __restrict__ b_out,
                      float* __restrict__ pre_pair)
{
    __shared__ __align__(16) float xn[32 * XPAD];     // layernormed chunk (32 s x 256 m)
    __shared__ __align__(16) float wab[CC * XPAD];    // w_ab staged (32 c x 256 m)
    __shared__ __align__(16) float atile[32 * APAD];  // a chunk (32 s x 32 c)
    __shared__ __align__(16) float oflat[CC * CC];    // accumulated outer product

    const int r    = blockIdx.x;
    const int tid  = threadIdx.x;
    const int wid  = tid >> 5;       // 4 waves
    const int lane = tid & 31;
    const int hi   = lane >> 4;      // half-wave select for WMMA fragments
    const int lx   = lane & 15;
    const int ti   = wid >> 1;       // output tile row (16-block)
    const int tj   = wid & 1;        // output tile col (16-block)

    // ---- stage w_ab into padded LDS (float4 / b128 path) ----
    {
        const float4* w4 = reinterpret_cast<const float4*>(w_ab);
        for (int idx = tid; idx < (CC * CM) / 4; idx += 128) {
            int row = idx >> 6;      // 64 float4 per row
            int c4  = idx & 63;
            *reinterpret_cast<float4*>(&wab[row * XPAD + c4 * 4]) = w4[idx];
        }
    }

    // per-lane gamma/beta cache: lane owns channels [4*lane..4*lane+3] and +128
    const float4* nw4 = reinterpret_cast<const float4*>(norm_w);
    const float4* nb4 = reinterpret_cast<const float4*>(norm_b);
    const float4 g0 = nw4[lane],     g1 = nw4[lane + 32];
    const float4 be0 = nb4[lane],    be1 = nb4[lane + 32];

    // bias for GEMM1 C-init: column n = tj*16+lx, identical in all 8 D rows
    const float bias = b_ab[tj * 16 + lx];
    v8f bias8;
    #pragma unroll
    for (int i = 0; i < 8; ++i) bias8[i] = bias;

    v8f oacc = {};   // this wave's 16x16 tile of the 32x32 accumulated A^T A

    #pragma unroll 1
    for (int chunk = 0; chunk < NSEQ / 32; ++chunk) {
        const int sbase = chunk * 32;

        // ---- load 8 s-rows per wave (coalesced b128) + LayerNorm ----
        #pragma unroll 1
        for (int q = 0; q < 8; ++q) {
            const int rr = wid * 8 + q;
            const float4* src4 = reinterpret_cast<const float4*>(
                msa + ((size_t)(sbase + rr) * NRES + r) * CM);
            float4 x0 = src4[lane];
            float4 x1 = src4[lane + 32];
            if (chunk < NSEQ / 32 - 1) {  // prefetch next chunk's row
                const char* nxt = reinterpret_cast<const char*>(
                    msa + ((size_t)(sbase + 32 + rr) * NRES + r) * CM);
                __builtin_prefetch(nxt + lane * 16, 0, 1);
                __builtin_prefetch(nxt + (lane + 32) * 16, 0, 1);
            }
            float sum = x0.x + x0.y + x0.z + x0.w + x1.x + x1.y + x1.z + x1.w;
            float ssq = x0.x*x0.x + x0.y*x0.y + x0.z*x0.z + x0.w*x0.w
                      + x1.x*x1.x + x1.y*x1.y + x1.z*x1.z + x1.w*x1.w;
            #pragma unroll
            for (int off = 16; off > 0; off >>= 1) {
                sum += __shfl_xor(sum, off);
                ssq += __shfl_xor(ssq, off);
            }
            const float mu   = sum * (1.0f / CM);
            const float rstd = rsqrtf(ssq * (1.0f / CM) - mu * mu + 1e-5f);
            float4 y0, y1;
            y0.x = (x0.x - mu) * rstd * g0.x + be0.x;
            y0.y = (x0.y - mu) * rstd * g0.y + be0.y;
            y0.z = (x0.z - mu) * rstd * g0.z + be0.z;
            y0.w = (x0.w - mu) * rstd * g0.w + be0.w;
            y1.x = (x1.x - mu) * rstd * g1.x + be1.x;
            y1.y = (x1.y - mu) * rstd * g1.y + be1.y;
            y1.z = (x1.z - mu) * rstd * g1.z + be1.z;
            y1.w = (x1.w - mu) * rstd * g1.w + be1.w;
            float4* dst4 = reinterpret_cast<float4*>(&xn[rr * XPAD]);
            dst4[lane]      = y0;
            dst4[lane + 32] = y1;
        }
        __syncthreads();  // xn ready; also fences prior GEMM2's atile reads

        // ---- GEMM1: A(32s x 256m) x w_ab^T(256m x 32c), one 16x16 tile/wave ----
        // A frag (16x4 f32): lane lx holds M=lx; VGPR0=K(k0+2hi), VGPR1=K+1.
        // B frag (4x16):     lane lx holds N=lx; same K interleave.
        v8f acc = bias8;
        const int mrow = ti * 16 + lx;   // s row
        const int ncol = tj * 16 + lx;   // c col
        #pragma unroll 4
        for (int k0 = 0; k0 < CM; k0 += 4) {
            v2f af = *reinterpret_cast<const v2f*>(&xn[mrow * XPAD + k0 + 2 * hi]);
            v2f bf = *reinterpret_cast<const v2f*>(&wab[ncol * XPAD + k0 + 2 * hi]);
            acc = __builtin_amdgcn_wmma_f32_16x16x4_f32(
                false, af, false, bf, (short)0, acc, false, false);
        }
        // D layout: row = vgpr + 8*hi, col = lx
        #pragma unroll
        for (int vg = 0; vg < 8; ++vg)
            atile[(ti * 16 + vg + 8 * hi) * APAD + tj * 16 + lx] = acc[vg];
        __syncthreads();  // atile ready; fences GEMM1's xn/wab reads

        // ---- GEMM2: o += A^T A over this 32-s chunk (K=32) ----
        #pragma unroll
        for (int k0 = 0; k0 < 32; k0 += 4) {
            const int k = k0 + 2 * hi;
            v2f af, bf;
            af.x = atile[k * APAD + ti * 16 + lx];        // A^T[ci][k] = a[k][ci]
            af.y = atile[(k + 1) * APAD + ti * 16 + lx];
            bf.x = atile[k * APAD + tj * 16 + lx];        // B[k][cj]  = a[k][cj]
            bf.y = atile[(k + 1) * APAD + tj * 16 + lx];
            oacc = __builtin_amdgcn_wmma_f32_16x16x4_f32(
                false, af, false, bf, (short)0, oacc, false, false);
        }
        // next iteration's xn writes are fenced by the second __syncthreads, its atile writes by the first
    }

    // ---- flatten o (scaled by 1/s) into LDS ----
    #pragma unroll
    for (int vg = 0; vg < 8; ++vg)
        oflat[(ti * 16 + vg + 8 * hi) * CC + tj * 16 + lx] = oacc[vg] * (1.0f / NSEQ);
    __syncthreads();

    // ---- pre_pair[r][z] = o_flat . w_out[z,:] + b_out[z]; one z per thread ----
    {
        const int z = tid;  // 128 threads == CZ
        const float4* wrow = reinterpret_cast<const float4*>(w_out + (size_t)z * (CC * CC));
        const float4* o4   = reinterpret_cast<const float4*>(oflat);
        float acc = b_out[z];
        #pragma unroll 8
        for (int k = 0; k < (CC * CC) / 4; ++k) {
            const float4 wv = wrow[k];
            const float4 ov = o4[k];
            acc += wv.x * ov.x + wv.y * ov.y + wv.z * ov.z + wv.w * ov.w;
        }
        pre_pair[(size_t)r * CZ + z] = acc;
    }
}

// Broadcast pre_pair[r1,:] across r2: 512MB of pure streaming stores.
// grid (1024, 8) x 256 threads; each wave stores 512B contiguous, non-temporal.
// Uses clang native ext-vector (v4f) so __builtin_nontemporal_store accepts it.
__global__ __launch_bounds__(256)
void opm_broadcast_kernel(const float* __restrict__ pre_pair,
                          float* __restrict__ out)
{
    const int r1  = blockIdx.x;
    const int g   = blockIdx.y;          // 128-row slab of r2
    const int t   = threadIdx.x;
    const int z4  = t & 31;              // float4 index within the 128-ch row
    const int row = t >> 5;              // 0..7

    const v4f v = reinterpret_cast<const v4f*>(pre_pair + (size_t)r1 * CZ)[z4];
    v4f* dst = reinterpret_cast<v4f*>(out);
    const size_t base = ((size_t)r1 * NRES + (size_t)g * 128) * (CZ / 4);
    #pragma unroll
    for (int i = 0; i < 16; ++i) {
        __builtin_nontemporal_store(
            v, &dst[base + (size_t)(i * 8 + row) * (CZ / 4) + z4]);
    }
}

extern "C" void kernel_launch(void* const* d_in, const int* in_sizes, int n_in,
                              void* d_out, int out_size, void* d_ws, size_t ws_size,
                              hipStream_t stream) {
    const float* msa    = (const float*)d_in[0];
    const float* norm_w = (const float*)d_in[1];
    const float* norm_b = (const float*)d_in[2];
    const float* w_ab   = (const float*)d_in[3];
    const float* b_ab   = (const float*)d_in[4];
    const float* w_out  = (const float*)d_in[5];
    const float* b_out  = (const float*)d_in[6];
    float* out = (float*)d_out;
    float* pre_pair = (float*)d_ws;   // 1024*128 f32 = 512 KB scratch

    opm_fused_kernel<<<NRES, 128, 0, stream>>>(
        msa, norm_w, norm_b, w_ab, b_ab, w_out, b_out, pre_pair);
    opm_broadcast_kernel<<<dim3(NRES, 8), 256, 0, stream>>>(pre_pair, out);
}